// DCTLayer_46119358824466
// MI455X (gfx1250) — compile-verified
//
#include <hip/hip_runtime.h>

typedef __attribute__((ext_vector_type(2))) float v2f;
typedef __attribute__((ext_vector_type(4))) float v4f;
typedef __attribute__((ext_vector_type(8))) float v8f;

#define IMG_W     512
#define IMG_PIX   (512*512)
#define TILES_IMG (32*32)
#define N_TILES   (32*3*32*32)   /* 98304 16x16 tiles */
#define BLOCKS    2048
#define THREADS   256            /* 8 wave32 per block */

__device__ __forceinline__ void wave_lds_fence() {
    // Same-wave DS ops are in-order in hardware (DScnt); we only need to stop
    // the compiler from reordering LDS stores/loads across this point.
    __asm__ __volatile__("" ::: "memory");
    __builtin_amdgcn_wave_barrier();
    __asm__ __volatile__("" ::: "memory");
}

__global__ __launch_bounds__(THREADS)
void dct8x8_wmma_kernel(const float* __restrict__ x,
                        const float* __restrict__ dct,
                        float* __restrict__ out)
{
    __shared__ float lds[THREADS/32][256];   // 1 KB staging per wave

    const int lane = threadIdx.x & 31;
    const int wv   = threadIdx.x >> 5;
    const int half = lane >> 4;              // K-half for A/B operand layouts
    const int n    = lane & 15;              // M (A-op) / N (B-op) index

    // Ddiag = blockdiag(D, D), D = 8x8 dct basis.
    // a[k] = Ddiag[n, 4k + 2*half + {0,1}] in WMMA operand layout.
    // Serves BOTH as A-operand of W1 = Ddiag*Tile and as B-operand (Ddiag^T)
    // of M = W1*Ddiag^T (the two layouts coincide for this value pattern).
    v2f a[4];
#pragma unroll
    for (int k = 0; k < 4; ++k) {
        const int c    = 4*k + 2*half;
        const float* p = dct + (n & 7)*8 + (c & 7);   // always in-bounds
        const bool on  = ((n >> 3) == (c >> 3));      // block-diagonal mask
        a[k].x = on ? p[0] : 0.0f;
        a[k].y = on ? p[1] : 0.0f;
    }

    float* myLds = lds[wv];
    const int waveId = blockIdx.x * (THREADS >> 5) + wv;
    const int nWaves = BLOCKS * (THREADS >> 5);

    for (int t = waveId; t < N_TILES; t += nWaves) {   // uniform trip count
        const int img = t / TILES_IMG;
        const int rem = t - img*TILES_IMG;
        const int ty  = rem >> 5;
        const int tx  = rem & 31;
        const size_t base = (size_t)img*IMG_PIX + (size_t)ty*16*IMG_W + (size_t)tx*16;

        // ---- W1 = Ddiag * Tile : 4 chained V_WMMA_F32_16X16X4_F32 ----
        // B-op slice k: lane holds Tile[4k + 2*half + {0,1}, n]
        v8f acc1 = {0.f,0.f,0.f,0.f,0.f,0.f,0.f,0.f};
#pragma unroll
        for (int k = 0; k < 4; ++k) {
            const int row = 4*k + 2*half;
            v2f b;
            b.x = x[base + (size_t)row*IMG_W + n];
            b.y = x[base + (size_t)(row+1)*IMG_W + n];
            acc1 = __builtin_amdgcn_wmma_f32_16x16x4_f32(
                false, a[k], false, b, (short)0, acc1, false, false);
        }

        // ---- stage W1 row-major in wave-private LDS ----
        // C-layout: acc1[r] = W1[r + 8*half, n]
#pragma unroll
        for (int r = 0; r < 8; ++r)
            myLds[(r + 8*half)*16 + n] = acc1[r];
        wave_lds_fence();

        // ---- M = W1 * Ddiag^T : A-op = W1 rows (b64 LDS reads), B-op = a[k]
        v8f acc2 = {0.f,0.f,0.f,0.f,0.f,0.f,0.f,0.f};
#pragma unroll
        for (int k = 0; k < 4; ++k) {
            const int idx = n*16 + 4*k + 2*half;   // contiguous pair
            v2f aw;
            aw.x = myLds[idx];
            aw.y = myLds[idx + 1];
            acc2 = __builtin_amdgcn_wmma_f32_16x16x4_f32(
                false, aw, false, a[k], (short)0, acc2, false, false);
        }
        wave_lds_fence();   // order next iteration's LDS stores after these loads

        // ---- out block(i,j) = M block(i,j)^T, written in place ----
        // M element (m = rr + 8*half, n) -> tile pos (8*half + n%8, 8*(n/8) + rr)
        // => each lane owns one contiguous 8-float output row: 2x b128 stores.
        const size_t o = base + (size_t)(8*half + (n & 7))*IMG_W + (size_t)(8*(n >> 3));
        v4f lo = {acc2[0], acc2[1], acc2[2], acc2[3]};
        v4f hi = {acc2[4], acc2[5], acc2[6], acc2[7]};
        *(v4f*)(out + o)     = lo;
        *(v4f*)(out + o + 4) = hi;
    }
}

extern "C" void kernel_launch(void* const* d_in, const int* in_sizes, int n_in,
                              void* d_out, int out_size, void* d_ws, size_t ws_size,
                              hipStream_t stream) {
    const float* x   = (const float*)d_in[0];   // (32,3,512,512) f32
    const float* dct = (const float*)d_in[1];   // (8,8) f32
    float* out = (float*)d_out;                 // (32,3,512,512) f32
    dct8x8_wmma_kernel<<<BLOCKS, THREADS, 0, stream>>>(x, dct, out);
}